// SplitTopK_62594853372436
// MI455X (gfx1250) — compile-verified
//
#include <hip/hip_runtime.h>
#include <stdint.h>

// ---------------- problem constants ----------------
#define THREADS 256              // 8 wave32 per block
#define PART    16384            // elements per partition-row
#define KSEL    32               // top-k
#define CHUNK   4096             // floats per LDS staging chunk (16 KB)
#define NCHUNK  (PART / CHUNK)   // 4
#define CAP     2048             // candidate list capacity
#define V4_PER_CHUNK (CHUNK / 4 / THREADS)  // 4 float4 per thread per chunk
#define V4_PER_PART  (PART / 4 / THREADS)   // 16 float4 per thread per row
// sortable key of 2.0f: (0x40000000 | 0x80000000). >=32 elements of a 16384
// N(0,1) row exceed 2.0 with overwhelming probability; fallback covers the rest.
#define CKEY 0xC0000000u

typedef float v4f __attribute__((ext_vector_type(4)));
typedef int   v4i __attribute__((ext_vector_type(4)));
// pointer types matching the async-to-LDS builtin's signature:
//   (int4 addrspace(1)* src, int4 addrspace(3)* dst, imm offset, imm cpol)
typedef __attribute__((address_space(1))) v4i* g_v4i_ptr;
typedef __attribute__((address_space(3))) v4i* l_v4i_ptr;

#ifndef __has_builtin
#define __has_builtin(x) 0
#endif

#if __has_builtin(__builtin_amdgcn_global_load_async_to_lds_b128)
#define HAVE_ASYNC_LDS 1
#else
#define HAVE_ASYNC_LDS 0
#endif

// monotone float->uint key: larger float => larger unsigned key
__device__ __forceinline__ unsigned key_of(float f) {
  unsigned u = __float_as_uint(f);
  return (u & 0x80000000u) ? ~u : (u | 0x80000000u);
}
__device__ __forceinline__ float val_of(unsigned k) {
  unsigned u = (k & 0x80000000u) ? (k & 0x7fffffffu) : ~k;
  return __uint_as_float(u);
}

template <int N>
__device__ __forceinline__ void wait_async() {
#if HAVE_ASYNC_LDS
#if __has_builtin(__builtin_amdgcn_s_wait_asynccnt)
  __builtin_amdgcn_s_wait_asynccnt(N);
#else
  asm volatile("s_wait_asynccnt %0" ::"n"(N) : "memory");
#endif
#endif
}

__device__ __forceinline__ void wait_store0() {
#if __has_builtin(__builtin_amdgcn_s_wait_storecnt)
  __builtin_amdgcn_s_wait_storecnt(0);
#else
  __threadfence();
#endif
}

// copy one 16 KB chunk global -> LDS, async (VGPR-bypassing) when available
__device__ __forceinline__ void issue_chunk(const float* __restrict__ g,
                                            float* sbuf, int tid) {
#pragma unroll
  for (int i = 0; i < V4_PER_CHUNK; ++i) {
    const int off4 = (i * THREADS + tid) * 4;
#if HAVE_ASYNC_LDS
    __builtin_amdgcn_global_load_async_to_lds_b128(
        (g_v4i_ptr)(g + off4),
        (l_v4i_ptr)(sbuf + off4),
        /*offset=*/0, /*cpol=*/0);
#else
    *(v4f*)(sbuf + off4) = *(const v4f*)(g + off4);
#endif
  }
}

// thread-0 scan of a 256-bin histogram, high bins first; picks the bin that
// contains the `need`-th largest and the residual rank within it
__device__ __forceinline__ void select_bin(const unsigned* hist, unsigned* s_ctrl,
                                           unsigned prefix, unsigned need, int shift) {
  unsigned cum = 0u, chosen = 0u;
  for (int b = 255; b >= 0; --b) {
    const unsigned cnt = hist[b];
    if (cum + cnt >= need) { chosen = (unsigned)b; break; }
    cum += cnt;
  }
  s_ctrl[1] = prefix | (chosen << shift);
  s_ctrl[2] = need - cum;
}

extern "C" __global__ void __launch_bounds__(THREADS)
splittopk_kernel(const float* __restrict__ x, float* __restrict__ out) {
  const int tid = threadIdx.x;
  const long long base = (long long)blockIdx.x * PART;
  const float* __restrict__ src = x + base;
  float* __restrict__ dst = out + base;

  __shared__ float sbuf[2][CHUNK];           // 32 KB double buffer
  __shared__ unsigned cand_key[CAP];         // 8 KB
  __shared__ unsigned short cand_idx[CAP];   // 4 KB
  __shared__ unsigned hist[THREADS];         // 256 bins, 1 KB
  __shared__ unsigned s_ctrl[4];             // 0:nc 1:prefix 2:need 3:eq

  if (tid == 0) { s_ctrl[0] = 0u; s_ctrl[3] = 0u; }

  // kick off the async pipeline for chunks 0 and 1
  issue_chunk(src + 0 * CHUNK, sbuf[0], tid);
  issue_chunk(src + 1 * CHUNK, sbuf[1], tid);

  // stream zeros to the full output row while the copies are in flight
  {
    v4f z = {0.0f, 0.0f, 0.0f, 0.0f};
#pragma unroll
    for (int i = 0; i < V4_PER_PART; ++i) {
      const int off4 = (i * THREADS + tid) * 4;
      __builtin_nontemporal_store(z, (v4f*)(dst + off4));
    }
  }

  // consume chunks from LDS, compacting tail candidates (key >= CKEY)
#pragma unroll
  for (int c = 0; c < NCHUNK; ++c) {
    if (c < NCHUNK - 1) wait_async<V4_PER_CHUNK>(); else wait_async<0>();
    __syncthreads();                       // whole block's chunk c resident
    const float* sb = sbuf[c & 1];
#pragma unroll
    for (int i = 0; i < V4_PER_CHUNK; ++i) {
      const int off4 = (i * THREADS + tid) * 4;
      v4f v = *(const v4f*)(sb + off4);
#pragma unroll
      for (int e = 0; e < 4; ++e) {
        const unsigned k = key_of(v[e]);
        if (k >= CKEY) {
          const unsigned pos = atomicAdd(&s_ctrl[0], 1u);
          if (pos < CAP) {
            cand_key[pos] = k;
            cand_idx[pos] = (unsigned short)(c * CHUNK + off4 + e);
          }
        }
      }
    }
    __syncthreads();                       // everyone done reading this buffer
    if (c + 2 < NCHUNK) issue_chunk(src + (c + 2) * CHUNK, sbuf[c & 1], tid);
  }

  const unsigned nc = s_ctrl[0];
  unsigned prefix = 0u, need = KSEL;

  if (nc >= KSEL && nc <= CAP) {
    // ---- fast path: exact 4x8-bit radix select over the candidate list ----
    for (int level = 0; level < 4; ++level) {
      const int shift = 24 - 8 * level;
      const unsigned mhi = level ? (0xffffffffu << (32 - 8 * level)) : 0u;
      hist[tid] = 0u;
      __syncthreads();
      for (int j = tid; j < (int)nc; j += THREADS) {
        const unsigned k = cand_key[j];
        if ((k & mhi) == prefix) atomicAdd(&hist[(k >> shift) & 255u], 1u);
      }
      __syncthreads();
      if (tid == 0) select_bin(hist, s_ctrl, prefix, need, shift);
      __syncthreads();
      prefix = s_ctrl[1];
      need   = s_ctrl[2];
    }
    const unsigned T = prefix;             // exact key of 32nd largest
    wait_store0();                         // zero-fill globally visible...
    __syncthreads();                       // ...for every wave in the block
    for (int j = tid; j < (int)nc; j += THREADS) {
      const unsigned k = cand_key[j];
      if (k > T) {
        dst[cand_idx[j]] = fmaxf(val_of(k), 0.0f);
      } else if (k == T) {
        if (atomicAdd(&s_ctrl[3], 1u) < need)
          dst[cand_idx[j]] = fmaxf(val_of(k), 0.0f);
      }
    }
  } else {
    // ---- rare fallback: exact radix select over the full row (global) ----
    for (int level = 0; level < 4; ++level) {
      const int shift = 24 - 8 * level;
      const unsigned mhi = level ? (0xffffffffu << (32 - 8 * level)) : 0u;
      hist[tid] = 0u;
      __syncthreads();
      for (int i = 0; i < V4_PER_PART; ++i) {
        const int off4 = (i * THREADS + tid) * 4;
        v4f v = *(const v4f*)(src + off4);
#pragma unroll
        for (int e = 0; e < 4; ++e) {
          const unsigned k = key_of(v[e]);
          if ((k & mhi) == prefix) atomicAdd(&hist[(k >> shift) & 255u], 1u);
        }
      }
      __syncthreads();
      if (tid == 0) select_bin(hist, s_ctrl, prefix, need, shift);
      __syncthreads();
      prefix = s_ctrl[1];
      need   = s_ctrl[2];
    }
    const unsigned T = prefix;
    wait_store0();
    __syncthreads();
    for (int i = 0; i < V4_PER_PART; ++i) {
      const int off4 = (i * THREADS + tid) * 4;
      v4f v = *(const v4f*)(src + off4);
#pragma unroll
      for (int e = 0; e < 4; ++e) {
        const unsigned k = key_of(v[e]);
        if (k > T) {
          dst[off4 + e] = fmaxf(v[e], 0.0f);
        } else if (k == T) {
          if (atomicAdd(&s_ctrl[3], 1u) < need) dst[off4 + e] = fmaxf(v[e], 0.0f);
        }
      }
    }
  }
}

extern "C" void kernel_launch(void* const* d_in, const int* in_sizes, int n_in,
                              void* d_out, int out_size, void* d_ws, size_t ws_size,
                              hipStream_t stream) {
  (void)n_in; (void)out_size; (void)d_ws; (void)ws_size;
  const float* x = (const float*)d_in[0];
  float* out = (float*)d_out;
  const int nblk = in_sizes[0] / PART;   // 4096 rows * 2 partitions = 8192
  splittopk_kernel<<<nblk, THREADS, 0, stream>>>(x, out);
}